// HybridCNNQNN_78082505441470
// MI455X (gfx1250) — compile-verified
//
#include <hip/hip_runtime.h>
#include <math.h>

#define B_ 1024

typedef float v2f __attribute__((ext_vector_type(2)));
typedef float v8f __attribute__((ext_vector_type(8)));

// ---------------- conv1 (1->6, 5x5) + ReLU + 2x2 maxpool : (B,1,28,28) -> (B,6,12,12)
__global__ void k_conv1_pool(const float* __restrict__ x,
                             const float* __restrict__ w,
                             const float* __restrict__ bias,
                             float* __restrict__ h1) {
  int idx = blockIdx.x * blockDim.x + threadIdx.x;
  if (idx >= B_ * 6 * 144) return;
  int b  = idx / (6 * 144);
  int r  = idx % (6 * 144);
  int c  = r / 144;
  int p  = r % 144;
  int py = p / 12, px = p % 12;
  const float* in = x + b * 784;
  int iy0 = 2 * py, ix0 = 2 * px;

  float patch[36];
  #pragma unroll
  for (int i = 0; i < 6; i++)
    #pragma unroll
    for (int j = 0; j < 6; j++)
      patch[i * 6 + j] = in[(iy0 + i) * 28 + (ix0 + j)];

  float wv[25];
  #pragma unroll
  for (int i = 0; i < 25; i++) wv[i] = w[c * 25 + i];
  float bc = bias[c];

  float mx = -1e30f;
  #pragma unroll
  for (int dy = 0; dy < 2; dy++)
    #pragma unroll
    for (int dx = 0; dx < 2; dx++) {
      float s = bc;
      #pragma unroll
      for (int i = 0; i < 5; i++)
        #pragma unroll
        for (int j = 0; j < 5; j++)
          s = fmaf(patch[(dy + i) * 6 + (dx + j)], wv[i * 5 + j], s);
      mx = fmaxf(mx, s);
    }
  h1[idx] = fmaxf(mx, 0.f);   // relu(max) == max(relu) (monotone)
}

// ---------------- conv2 (6->16, 5x5) + ReLU + 2x2 maxpool : (B,6,12,12) -> (B,256)
__global__ void k_conv2_pool(const float* __restrict__ h1,
                             const float* __restrict__ w,
                             const float* __restrict__ bias,
                             float* __restrict__ h2) {
  __shared__ float sW[2400];   // 16*6*25
  __shared__ float sH[864];    // 6*12*12
  __shared__ float sB[16];
  int b = blockIdx.x, tid = threadIdx.x;
  for (int i = tid; i < 2400; i += 256) sW[i] = w[i];
  for (int i = tid; i < 864; i += 256) sH[i] = h1[b * 864 + i];
  if (tid < 16) sB[tid] = bias[tid];
  __syncthreads();

  int co = tid >> 4;          // 0..15
  int p  = tid & 15;          // 0..15 -> (py,px)
  int py = p >> 2, px = p & 3;

  float mx = -1e30f;
  #pragma unroll
  for (int dy = 0; dy < 2; dy++)
    #pragma unroll
    for (int dx = 0; dx < 2; dx++) {
      int oy = 2 * py + dy, ox = 2 * px + dx;   // 0..7
      float s = sB[co];
      for (int ci = 0; ci < 6; ci++) {
        const float* hp = &sH[ci * 144 + oy * 12 + ox];
        const float* wp = &sW[(co * 6 + ci) * 25];
        #pragma unroll
        for (int i = 0; i < 5; i++)
          #pragma unroll
          for (int j = 0; j < 5; j++)
            s = fmaf(hp[i * 12 + j], wp[i * 5 + j], s);
      }
      mx = fmaxf(mx, s);
    }
  // flatten order matches reference reshape: k = co*16 + py*4 + px == tid
  h2[b * 256 + tid] = fmaxf(mx, 0.f);
}

// ---------------- FC1: (B,256) @ (120,256)^T + b, ReLU -- fp32 WMMA 16x16x4
// One wave (32 lanes) per 16x16 output tile; K=256 -> 64 WMMA ops.
__global__ void k_fc1_wmma(const float* __restrict__ h2,
                           const float* __restrict__ w,    // (120,256) row-major
                           const float* __restrict__ bias, // (120,)
                           float* __restrict__ h3) {       // (B,120)
  const int m0   = blockIdx.x * 16;
  const int n0   = blockIdx.y * 16;
  const int lane = threadIdx.x;      // 0..31, one full wave, EXEC all ones
  const int half = lane >> 4;        // 0: K pair {0,1}, 1: K pair {2,3}
  const int mr   = lane & 15;

  const int   n     = n0 + mr;
  const float bmask = (n < 120) ? 1.f : 0.f;       // mask instead of branch (keep EXEC full)
  const int   nsafe = (n < 120) ? n : 0;

  const float* arow = h2 + (size_t)(m0 + mr) * 256 + 2 * half;  // A[m][k]
  const float* brow = w  + (size_t)nsafe    * 256 + 2 * half;   // B[k][n] = W[n][k]

  v8f acc = {0.f, 0.f, 0.f, 0.f, 0.f, 0.f, 0.f, 0.f};
  #pragma unroll 8
  for (int k0 = 0; k0 < 256; k0 += 4) {
    v2f a, bb;
    a.x  = arow[k0];
    a.y  = arow[k0 + 1];
    bb.x = brow[k0] * bmask;
    bb.y = brow[k0 + 1] * bmask;
    // 8 args: (neg_a, A, neg_b, B, c_mod, C, reuse_a, reuse_b)
    acc = __builtin_amdgcn_wmma_f32_16x16x4_f32(false, a, false, bb,
                                                (short)0, acc, false, false);
  }

  if (n < 120) {
    float bv = bias[n];
    #pragma unroll
    for (int r = 0; r < 8; r++) {
      int row = m0 + half * 8 + r;            // C/D layout: VGPR r holds M=r / M=8+r
      h3[(size_t)row * 120 + n] = fmaxf(acc[r] + bv, 0.f);
    }
  }
}

// ---------------- FC2 + sigmoid + 10-qubit circuit, one block per batch element.
// State (1024 complex) lives entirely in LDS; deterministic reduction.
__global__ void k_quantum(const float* __restrict__ h3,  // (B,120)
                          const float* __restrict__ fw,  // (10,120)
                          const float* __restrict__ fb,  // (10,)
                          const float* __restrict__ qw,  // (2,10,3)
                          float* __restrict__ out) {     // (B,10)
  __shared__ float sre[1024];
  __shared__ float sim[1024];
  __shared__ float ang[10];
  __shared__ float zbuf[10 * 256];
  const int b = blockIdx.x, tid = threadIdx.x;

  // FC2 -> angles = pi * sigmoid(feats)
  if (tid < 10) {
    const float* hr = h3 + (size_t)b * 120;
    const float* wr = fw + (size_t)tid * 120;
    float s = fb[tid];
    for (int k = 0; k < 120; k++) s = fmaf(hr[k], wr[k], s);
    ang[tid] = 3.14159265358979f / (1.f + __expf(-s));
  }
  for (int i = tid; i < 1024; i += 256) {
    sre[i] = (i == 0) ? 1.f : 0.f;
    sim[i] = 0.f;
  }
  __syncthreads();

  // AngleEmbedding: RX(a) on each wire. Qubit q <-> bit (9-q) of flat index.
  for (int q = 0; q < 10; q++) {
    float ha = 0.5f * ang[q];
    float c = __cosf(ha), s = __sinf(ha);
    int bp = 9 - q, st = 1 << bp;
    for (int p = tid; p < 512; p += 256) {
      int i0 = ((p >> bp) << (bp + 1)) | (p & (st - 1));
      int i1 = i0 | st;
      float r0 = sre[i0], m0 = sim[i0], r1 = sre[i1], m1 = sim[i1];
      // RX: [[c, -i s],[-i s, c]]
      sre[i0] = fmaf(c, r0,  s * m1);  sim[i0] = fmaf(c, m0, -s * r1);
      sre[i1] = fmaf(s, m0,  c * r1);  sim[i1] = fmaf(-s, r0, c * m1);
    }
    __syncthreads();
  }

  for (int l = 0; l < 2; l++) {
    // Rot(phi, theta, omega) = RZ(omega) RY(theta) RZ(phi) on each wire
    for (int q = 0; q < 10; q++) {
      float phi = qw[(l * 10 + q) * 3 + 0];
      float th  = qw[(l * 10 + q) * 3 + 1];
      float om  = qw[(l * 10 + q) * 3 + 2];
      float ct = __cosf(0.5f * th), stn = __sinf(0.5f * th);
      float ap = 0.5f * (phi + om), am = 0.5f * (phi - om);
      float cap = __cosf(ap), sap = __sinf(ap);
      float cam = __cosf(am), sam = __sinf(am);
      float u00r =  ct * cap, u00i = -ct * sap;   // e^{-i(phi+om)/2} cos
      float u01r = -stn * cam, u01i = -stn * sam; // -e^{ i(phi-om)/2} sin
      float u10r =  stn * cam, u10i = -stn * sam; //  e^{-i(phi-om)/2} sin
      float u11r =  ct * cap, u11i =  ct * sap;   //  e^{ i(phi+om)/2} cos
      int bp = 9 - q, st = 1 << bp;
      for (int p = tid; p < 512; p += 256) {
        int i0 = ((p >> bp) << (bp + 1)) | (p & (st - 1));
        int i1 = i0 | st;
        float r0 = sre[i0], m0 = sim[i0], r1 = sre[i1], m1 = sim[i1];
        sre[i0] = u00r * r0 - u00i * m0 + u01r * r1 - u01i * m1;
        sim[i0] = u00r * m0 + u00i * r0 + u01r * m1 + u01i * r1;
        sre[i1] = u10r * r0 - u10i * m0 + u11r * r1 - u11i * m1;
        sim[i1] = u10r * m0 + u10i * r0 + u11r * m1 + u11i * r1;
      }
      __syncthreads();
    }
    // CNOT ring with range r = (l % 9) + 1
    int rr = (l % 9) + 1;
    for (int q = 0; q < 10; q++) {
      int cb = 9 - q;
      int tb = 9 - ((q + rr) % 10);
      int hi = cb > tb ? cb : tb;
      int lo = cb > tb ? tb : cb;
      int p = tid;                     // 256 disjoint pairs, one per thread
      int t1 = ((p >> lo) << (lo + 1)) | (p & ((1 << lo) - 1));
      int t2 = ((t1 >> hi) << (hi + 1)) | (t1 & ((1 << hi) - 1));
      int i0 = t2 | (1 << cb);         // ctrl=1, tgt=0
      int i1 = i0 | (1 << tb);         // ctrl=1, tgt=1
      float r0 = sre[i0], m0 = sim[i0];
      float r1 = sre[i1], m1 = sim[i1];
      sre[i0] = r1; sim[i0] = m1;
      sre[i1] = r0; sim[i1] = m0;
      __syncthreads();
    }
  }

  // <Z_q> = sum_s |amp(s)|^2 * (1 - 2*bit_q(s)); deterministic two-stage reduce
  float z[10];
  #pragma unroll
  for (int q = 0; q < 10; q++) z[q] = 0.f;
  for (int k = 0; k < 4; k++) {
    int idx = tid + 256 * k;
    float pr = sre[idx] * sre[idx] + sim[idx] * sim[idx];
    #pragma unroll
    for (int q = 0; q < 10; q++)
      z[q] += ((idx >> (9 - q)) & 1) ? -pr : pr;
  }
  #pragma unroll
  for (int q = 0; q < 10; q++) zbuf[q * 256 + tid] = z[q];
  __syncthreads();
  if (tid < 10) {
    float s = 0.f;
    for (int t = 0; t < 256; t++) s += zbuf[tid * 256 + t];
    out[(size_t)b * 10 + tid] = s;
  }
}

extern "C" void kernel_launch(void* const* d_in, const int* in_sizes, int n_in,
                              void* d_out, int out_size, void* d_ws, size_t ws_size,
                              hipStream_t stream) {
  const float* x       = (const float*)d_in[0];
  const float* conv1_w = (const float*)d_in[1];
  const float* conv1_b = (const float*)d_in[2];
  const float* conv2_w = (const float*)d_in[3];
  const float* conv2_b = (const float*)d_in[4];
  const float* fc1_w   = (const float*)d_in[5];
  const float* fc1_b   = (const float*)d_in[6];
  const float* fc2_w   = (const float*)d_in[7];
  const float* fc2_b   = (const float*)d_in[8];
  const float* qw      = (const float*)d_in[9];
  float* out = (float*)d_out;

  float* h1 = (float*)d_ws;                    // B*6*12*12 = 884736 floats
  float* h2 = h1 + (size_t)B_ * 864;           // B*256     = 262144 floats
  float* h3 = h2 + (size_t)B_ * 256;           // B*120     = 122880 floats

  k_conv1_pool<<<(B_ * 6 * 144 + 255) / 256, 256, 0, stream>>>(x, conv1_w, conv1_b, h1);
  k_conv2_pool<<<B_, 256, 0, stream>>>(h1, conv2_w, conv2_b, h2);
  k_fc1_wmma<<<dim3(B_ / 16, 8), 32, 0, stream>>>(h2, fc1_w, fc1_b, h3);
  k_quantum<<<B_, 256, 0, stream>>>(h3, fc2_w, fc2_b, qw, out);
}